// AttentionOptimizedNeuron_84155589198569
// MI455X (gfx1250) — compile-verified
//
#include <hip/hip_runtime.h>
#include <stdint.h>

// ---------------------------------------------------------------------------
// Attention-modulated LIF neuron scan, gfx1250.
//   att = a_att*att + aw*0.2 ; mem = a_mem*mem + x*(1+att)
//   s = (mem >= 1) ; mem -= s
// Memory-bound: 384 MB @ 23.3 TB/s ~ 16.5 us floor. One lane per (b,d) chain
// (32768 chains -> 1024 waves). 48-deep async global->LDS pipeline per wave
// (ASYNCcnt flow control, wave-private LDS rings, zero barriers) to keep
// ~12 MB of HBM traffic in flight (Little's law vs ~1 us latency).
// ---------------------------------------------------------------------------

namespace {
constexpr int   kT       = 1024;
constexpr int   kD       = 1024;
constexpr int   kStages  = 48;          // async pipeline depth (<= 63 ASYNCcnt)
constexpr int   kBlock   = 64;          // 2 independent waves per workgroup
constexpr int   kWaves   = kBlock / 32;
constexpr float kAlphaMem = 0.95122942450071400910f;  // exp(-1/20)
constexpr float kAlphaAtt = 0.98019867330675525309f;  // exp(-1/50)
constexpr float kAttGain  = 0.2f;
constexpr float kThreshold = 1.0f;
}

// One wave moves each stage (64 x-floats + 64 aw-floats = 512B; aw at +128B) with a
// single async B128: lanes 0-15 carry x, lanes 16-31 carry aw (16B per lane).
#define ASYNC_B128_TO_LDS(ldsOff, gptr)                                        \
  asm volatile("global_load_async_to_lds_b128 %0, %1, off"                     \
               :: "v"(ldsOff), "v"(gptr) : "memory")

#define WAIT_ASYNCCNT(n)                                                       \
  asm volatile("s_wait_asynccnt %0" :: "n"(n) : "memory")

#define WAIT_DSCNT0()                                                          \
  asm volatile("s_wait_dscnt 0" ::: "memory")

__global__ __launch_bounds__(kBlock)
void snn_attn_scan(const float* __restrict__ x,
                   const float* __restrict__ aw,
                   float* __restrict__ out)
{
  // Per-wave ring: kStages stages x (32 x | 32 aw) floats = 12KB per wave.
  __shared__ __align__(16) float ring[kWaves][kStages][64];

  const int tid  = threadIdx.x;
  const int lane = tid & 31;
  const int wv   = tid >> 5;

  const int elem0 = blockIdx.x * kBlock + (wv << 5);  // wave's first chain
  const int elem  = elem0 + lane;                     // this lane's chain
  const int b     = elem >> 10;                       // elem / D
  const int d     = elem & (kD - 1);                  // elem % D
  const size_t colBase = (size_t)b * kT * kD + d;

  // --- async copy lane roles ---------------------------------------------
  const int q  = lane & 15;
  const int bw = elem0 >> 10;
  const int d0 = elem0 & (kD - 1);
  const float* gp = ((lane < 16) ? x : aw)
                    + ((size_t)bw * kT * kD + d0 + 4 * q);   // 16B per lane

  const uint32_t ringBase = (uint32_t)(size_t)&ring[wv][0][0];   // LDS offset
  const uint32_t copyOff  = ringBase + ((lane < 16) ? 0u : 128u)
                          + 16u * (uint32_t)q;

  // --- prologue: fill the pipeline ---------------------------------------
#pragma unroll
  for (int s = 0; s < kStages; ++s) {
    ASYNC_B128_TO_LDS(copyOff + (uint32_t)s * 256u, gp);
    gp += kD;
  }

  float mem = 0.0f, att = 0.0f;
  float*       op    = out + colBase;
  const float* xslot = &ring[wv][0][lane];

  int slot = 0;
  // --- steady state: wait oldest stage, compute, refill its slot ---------
  for (int t = 0; t < kT - kStages; ++t) {
    WAIT_ASYNCCNT(kStages - 1);            // oldest async load has landed
    const float xv = xslot[slot * 64];
    const float av = xslot[slot * 64 + 32];

    att = kAlphaAtt * att + av * kAttGain;
    mem = kAlphaMem * mem + xv * (1.0f + att);
    const float s = (mem >= kThreshold) ? 1.0f : 0.0f;
    mem -= s;                              // subtract reset (threshold == 1)
    __builtin_nontemporal_store(s, op);
    op += kD;

    WAIT_DSCNT0();                         // slot reads retired before reuse
    ASYNC_B128_TO_LDS(copyOff + (uint32_t)slot * 256u, gp);  // stage t+kStages
    gp += kD;
    slot = (slot + 1 == kStages) ? 0 : slot + 1;
  }

  // --- tail: drain once, last kStages stages are all resident in LDS -----
  WAIT_ASYNCCNT(0);
#pragma unroll 4
  for (int t = kT - kStages; t < kT; ++t) {
    const float xv = xslot[slot * 64];
    const float av = xslot[slot * 64 + 32];
    att = kAlphaAtt * att + av * kAttGain;
    mem = kAlphaMem * mem + xv * (1.0f + att);
    const float s = (mem >= kThreshold) ? 1.0f : 0.0f;
    mem -= s;
    __builtin_nontemporal_store(s, op);
    op += kD;
    slot = (slot + 1 == kStages) ? 0 : slot + 1;
  }
}

extern "C" void kernel_launch(void* const* d_in, const int* in_sizes, int n_in,
                              void* d_out, int out_size, void* d_ws, size_t ws_size,
                              hipStream_t stream) {
  (void)n_in; (void)out_size; (void)d_ws; (void)ws_size;
  const float* x  = (const float*)d_in[0];
  const float* aw = (const float*)d_in[1];
  float* out = (float*)d_out;

  const int total = in_sizes[0];       // B*T*D
  const int cols  = total / kT;        // B*D independent scan chains (32768)
  dim3 grid(cols / kBlock), block(kBlock);
  snn_attn_scan<<<grid, block, 0, stream>>>(x, aw, out);
}